// FewShotSegmentation_21268678050081
// MI455X (gfx1250) — compile-verified
//
#include <hip/hip_runtime.h>
#include <hip/hip_bf16.h>
#include <math.h>

// Problem constants (from reference): B=8, C=1024, H=W=64, P=4096, K=16
#define BATCH 8
#define CHAN  1024
#define PIX   4096
#define NCLS  16

typedef float v2f __attribute__((ext_vector_type(2)));
typedef float v8f __attribute__((ext_vector_type(8)));

// Interleaved prototype layout: element (c, k) -> [(c>>1)*32 + k*2 + (c&1)]
// Consecutive channel pairs for one class are adjacent => ds_load_b64 in match.
__device__ __forceinline__ int proto_idx(int c, int k) {
    return ((c >> 1) << 5) + (k << 1) + (c & 1);
}

// ---------------------------------------------------------------------------
// Kernel 1: per-batch class histogram (den) + zero pn2 accumulators
// ---------------------------------------------------------------------------
__global__ __launch_bounds__(256) void fss_hist(const int* __restrict__ mask,
                                                float* __restrict__ den,
                                                float* __restrict__ pn2) {
    __shared__ int h[NCLS];
    const int b = blockIdx.x;
    const int t = threadIdx.x;
    if (t < NCLS) h[t] = 0;
    __syncthreads();
    for (int p = t; p < PIX; p += 256) {
        int m = mask[b * PIX + p];
        if (m >= 1 && m <= NCLS) atomicAdd(&h[m - 1], 1);
    }
    __syncthreads();
    if (t < NCLS) {
        den[b * NCLS + t] = (float)h[t];
        pn2[b * NCLS + t] = 0.0f;
    }
}

// ---------------------------------------------------------------------------
// Kernel 2: prototypes via V_WMMA_F32_16X16X4_F32
//   D(16 chan x 16 cls) += A(16 chan x 4 pix) * B(4 pix x 16 cls one-hot)
// One wave per (batch, 16-channel tile). Stores prototypes in the pair-
// interleaved layout and accumulates pn2[b][k] = sum_c proto^2 via atomics.
// ---------------------------------------------------------------------------
__global__ __launch_bounds__(32) void fss_proto(const float* __restrict__ f,
                                                const int* __restrict__ mask,
                                                const float* __restrict__ den,
                                                float* __restrict__ protoI,
                                                float* __restrict__ pn2) {
    const int bb   = blockIdx.x >> 6;          // batch
    const int c0   = (blockIdx.x & 63) << 4;   // channel tile base
    const int lane = threadIdx.x;              // 0..31 (wave32)
    const int half = lane >> 4;                // 0: K=0,1   1: K=2,3
    const int l15  = lane & 15;
    const int cls  = l15 + 1;                  // foreground class id 1..16

    const float* frow = f + ((size_t)(bb * CHAN + c0 + l15)) * PIX;  // A row (M=l15)
    const int*   mrow = mask + (size_t)bb * PIX;

    v8f acc = {};
#pragma unroll 4
    for (int p = 0; p < PIX; p += 4) {
        const int pk = p + half * 2;
        // A fragment: M = l15 (channel), K = pk, pk+1 (pixels) -> b64 load
        float2 fa = *(const float2*)(frow + pk);
        // B fragment: one-hot mask rows K=pk, pk+1 over classes (N = l15)
        int m0 = mrow[pk];
        int m1 = mrow[pk + 1];
        v2f a;  a.x  = fa.x;                      a.y  = fa.y;
        v2f bv; bv.x = (m0 == cls) ? 1.0f : 0.0f; bv.y = (m1 == cls) ? 1.0f : 0.0f;
        acc = __builtin_amdgcn_wmma_f32_16x16x4_f32(false, a, false, bv,
                                                    (short)0, acc, false, false);
    }

    const float inv = 1.0f / den[bb * NCLS + l15];
    float s = 0.0f;
    float* pI = protoI + (size_t)bb * CHAN * NCLS;
#pragma unroll
    for (int r = 0; r < 8; ++r) {
        // D layout: VGPR r -> chan row r (lanes 0-15) / r+8 (lanes 16-31), N=l15
        float pr  = acc[r] * inv;
        int   row = c0 + r + (half << 3);
        pI[proto_idx(row, l15)] = pr;
        s += pr * pr;
    }
    s += __shfl_xor(s, 16, 32);                  // combine channel halves per class
    if (lane < 16) atomicAdd(&pn2[bb * NCLS + lane], s);
}

// ---------------------------------------------------------------------------
// Kernel 3: cosine match + argmax via V_WMMA_F32_16X16X4_F32
//   D(16 cls x 16 pix) += A(16 cls x 4 chan, LDS) * B(4 chan x 16 pix, global)
// Class-major D tile: lane n holds classes (half*8 .. half*8+7) for pixel
// p0+(n&15) => argmax = 8 VALU compares + one shfl_xor(16); output is one
// coalesced 64B store per tile. ||q|| accumulated for free from B fragments.
// ---------------------------------------------------------------------------
__global__ __launch_bounds__(128) void fss_match(const float* __restrict__ q,
                                                 const float* __restrict__ protoI,
                                                 const float* __restrict__ pn2,
                                                 float* __restrict__ out) {
    __shared__ float sp[CHAN * NCLS];            // 64 KB: interleaved prototypes
    const int bb  = blockIdx.x >> 4;             // batch
    const int blk = blockIdx.x & 15;             // 256-pixel slab within batch
    const int tid = threadIdx.x;

    // Cooperative 64KB LDS fill (float4, fully coalesced, layout-agnostic copy)
    const float4* src = (const float4*)(protoI + (size_t)bb * CHAN * NCLS);
    float4*       dst = (float4*)sp;
#pragma unroll
    for (int i = 0; i < (CHAN * NCLS / 4) / 128; ++i)
        dst[tid + i * 128] = src[tid + i * 128];
    __syncthreads();

    const int wave = tid >> 5;
    const int lane = tid & 31;
    const int half = lane >> 4;
    const int l15  = lane & 15;

    // Per-lane class norms: lanes<16 need classes 0-7, lanes>=16 classes 8-15
    float pnv[8];
#pragma unroll
    for (int r = 0; r < 8; ++r)
        pnv[r] = sqrtf(pn2[bb * NCLS + (half << 3) + r]);

    const float* qb = q + (size_t)bb * CHAN * PIX;

    for (int t = 0; t < 4; ++t) {
        const int p0 = blk * 256 + (wave * 4 + t) * 16;   // 16-pixel tile base
        v8f acc = {};
        float qsq = 0.0f;
#pragma unroll 4
        for (int cc = 0; cc < CHAN; cc += 4) {
            // A fragment (LDS): class l15, channels cc+half*2, +1 -> ds_load_b64
            float2 av = *(const float2*)(&sp[(((cc >> 1) + half) << 5) + (l15 << 1)]);
            // B fragment (global): channel rows cc+half*2, +1 at pixel p0+l15
            const int r0 = cc + half * 2;
            float b0 = qb[(size_t)r0 * PIX + p0 + l15];
            float b1 = qb[(size_t)(r0 + 1) * PIX + p0 + l15];
            qsq += b0 * b0 + b1 * b1;            // each (c,p) touched exactly once
            v2f a;  a.x  = av.x; a.y  = av.y;
            v2f bv; bv.x = b0;   bv.y = b1;
            acc = __builtin_amdgcn_wmma_f32_16x16x4_f32(false, a, false, bv,
                                                        (short)0, acc, false, false);
        }
        // Full ||q_{p0+l15}||: halves covered channels {0,1} vs {2,3} mod 4
        const float qn = sqrtf(qsq + __shfl_xor(qsq, 16, 32));

        // Per-lane argmax over this half's 8 classes (ascending r = first-max)
        float bbv = -__builtin_huge_valf();
        int   bi  = 0;
#pragma unroll
        for (int r = 0; r < 8; ++r) {
            float v = acc[r] / fmaxf(pnv[r] * qn, 1e-8f);
            if (v > bbv) { bbv = v; bi = (half << 3) + r; }
        }
        // Merge the two 8-class halves (tie -> smaller class index)
        float ov = __shfl_xor(bbv, 16, 32);
        int   oi = __shfl_xor(bi, 16, 32);
        if (ov > bbv || (ov == bbv && oi < bi)) { bbv = ov; bi = oi; }

        if (lane < 16)                            // one coalesced 64B store
            out[(size_t)bb * PIX + p0 + l15] = (float)bi;
    }
}

// ---------------------------------------------------------------------------
extern "C" void kernel_launch(void* const* d_in, const int* in_sizes, int n_in,
                              void* d_out, int out_size, void* d_ws, size_t ws_size,
                              hipStream_t stream) {
    (void)in_sizes; (void)n_in; (void)out_size; (void)ws_size;
    const float* sf = (const float*)d_in[0];   // support_features (B,C,H,W) f32
    const int*   sm = (const int*)d_in[1];     // support_masks    (B,1,H,W) i32
    const float* qf = (const float*)d_in[2];   // query_features   (B,C,H,W) f32
    float* out = (float*)d_out;                // (B,H,W) class indices as f32

    // Workspace: protoI (B*C*K f32 = 512KB) | den (128 f32) | pn2 (128 f32)
    float* protoI = (float*)d_ws;
    float* den    = protoI + (size_t)BATCH * CHAN * NCLS;
    float* pn2    = den + BATCH * NCLS;

    fss_hist <<<BATCH, 256, 0, stream>>>(sm, den, pn2);
    fss_proto<<<BATCH * (CHAN / 16), 32, 0, stream>>>(sf, sm, den, protoI, pn2);
    fss_match<<<BATCH * (PIX / 256), 128, 0, stream>>>(qf, protoI, pn2, out);
}